// ConvEncoderND_23776938951441
// MI455X (gfx1250) — compile-verified
//
#include <hip/hip_runtime.h>

typedef __attribute__((ext_vector_type(2))) float v2f;
typedef __attribute__((ext_vector_type(8))) float v8f;

#define GRID_N 64
#define NPTS   4096   // GRID_N*GRID_N
#define NCTX   512
#define NCH    64
#define LOG2E  1.44269504088896340736f

// One block = one batch slice of 256 output points.
// Phase 1: kernel-smoothing over 512 context points held in LDS
//          (raw v_sqrt_f32 / v_exp_f32 — arguments are range-safe).
// Phase 2: fused 2->64 channel projection via V_WMMA_F32_16X16X4_F32,
//          bias + sigmoid epilogue (v_rcp_f32), transposed (B,C,NPTS) store.
__global__ __launch_bounds__(256) void convnp_fused(
    const float* __restrict__ xc,    // (B,512,2)
    const float* __restrict__ yc,    // (B,512,1)
    const float* __restrict__ xg,    // (B,4096,2)
    const float* __restrict__ sig,   // (2,)
    const float* __restrict__ W,     // (64,2)
    const float* __restrict__ bias,  // (64,)
    float* __restrict__ out)         // (B,64,4096)
{
    __shared__ float sxc[NCTX * 2];
    __shared__ float syc[NCTX];
    __shared__ float sy[256 * 2];    // per-point (dens, conv/dens)

    const int t    = threadIdx.x;
    const int b    = blockIdx.x >> 4;   // 16 blocks per batch
    const int tile = blockIdx.x & 15;
    const int po   = tile << 8;         // first point of this block

    // ---- stage context for this batch into LDS ----
    const float* xcb = xc + (size_t)b * NCTX * 2;
    const float* ycb = yc + (size_t)b * NCTX;
    for (int i = t; i < NCTX; i += 256) {
        sxc[2 * i]     = xcb[2 * i];
        sxc[2 * i + 1] = xcb[2 * i + 1];
        syc[i]         = ycb[i];
    }
    __syncthreads();

    // per-channel exponents: exp(-0.5*d/s^2) == exp2(d * c)
    const float s0 = __expf(sig[0]);
    const float s1 = __expf(sig[1]);
    const float c0 = -0.5f * LOG2E / (s0 * s0);
    const float c1 = -0.5f * LOG2E / (s1 * s1);

    const int   o  = po + t;
    const float gx = xg[((size_t)b * NPTS + o) * 2 + 0];
    const float gy = xg[((size_t)b * NPTS + o) * 2 + 1];

    float dens = 0.f, conv = 0.f;
#pragma unroll 4
    for (int i = 0; i < NCTX; ++i) {
        float dx   = gx - sxc[2 * i];
        float dy   = gy - sxc[2 * i + 1];
        // raw v_sqrt_f32: argument is a small non-negative sum of squares
        float dist = __builtin_amdgcn_sqrtf(__builtin_fmaf(dx, dx, dy * dy));
        // raw v_exp_f32: dist*c in [-103, 0] for sigma=0.1 — no denorm guard needed
        float e0   = __builtin_amdgcn_exp2f(dist * c0);
        float e1   = __builtin_amdgcn_exp2f(dist * c1);
        dens += e0;
        conv  = __builtin_fmaf(syc[i], e1, conv);
    }
    sy[2 * t]     = dens;
    sy[2 * t + 1] = conv * __builtin_amdgcn_rcpf(dens + 1e-8f);
    __syncthreads();

    // ---- phase 2: WMMA projection ----
    // D[ch][pt] = sum_k W[ch][k] * y[pt][k]   (A = W tile 16x4, B = y tile 4x16)
    const int  wave  = t >> 5;
    const int  lane  = t & 31;
    const int  lhalf = lane & 15;
    const bool lo    = lane < 16;

    float* outb = out + (size_t)b * NCH * NPTS;

    for (int pt = 2 * wave; pt < 2 * wave + 2; ++pt) {
        // B operand (4x16): lanes 0-15 hold {y0,y1} for point N=lane; K=2,3 are zero.
        float by0 = sy[2 * (pt * 16 + lhalf)];
        float by1 = sy[2 * (pt * 16 + lhalf) + 1];
        v2f bm;
        bm.x = lo ? by0 : 0.f;
        bm.y = lo ? by1 : 0.f;

        const int oPt = po + pt * 16 + lhalf;

        for (int ct = 0; ct < 4; ++ct) {
            // A operand (16x4): lanes 0-15 hold {W[M][0], W[M][1]} for row M=lane.
            float aw0 = W[(ct * 16 + lhalf) * 2 + 0];
            float aw1 = W[(ct * 16 + lhalf) * 2 + 1];
            v2f am;
            am.x = lo ? aw0 : 0.f;
            am.y = lo ? aw1 : 0.f;

            v8f acc = {};
            acc = __builtin_amdgcn_wmma_f32_16x16x4_f32(
                /*neg_a=*/false, am, /*neg_b=*/false, bm,
                /*c_mod=*/(short)0, acc, /*reuse_a=*/false, /*reuse_b=*/false);

            // D layout: vgpr j, lanes 0-15 -> ch = ct*16+j;   lanes 16-31 -> ch = ct*16+8+j
            const int kBase = ct * 16 + (lo ? 0 : 8);
#pragma unroll
            for (int j = 0; j < 8; ++j) {
                float v = acc[j] + bias[kBase + j];
                // sigmoid via raw exp2 + rcp; v is a bounded pre-activation
                float r = __builtin_amdgcn_rcpf(
                    1.f + __builtin_amdgcn_exp2f(-LOG2E * v));
                outb[(size_t)(kBase + j) * NPTS + oPt] = r;  // two 64B coalesced segments
            }
        }
    }
}

extern "C" void kernel_launch(void* const* d_in, const int* in_sizes, int n_in,
                              void* d_out, int out_size, void* d_ws, size_t ws_size,
                              hipStream_t stream) {
    (void)n_in; (void)out_size; (void)d_ws; (void)ws_size;
    const float* xc   = (const float*)d_in[0];  // x_context (B,512,2)
    const float* yc   = (const float*)d_in[1];  // y_context (B,512,1)
    const float* xg   = (const float*)d_in[2];  // x_grid    (B,4096,2)
    const float* sig  = (const float*)d_in[3];  // sigma     (2,)
    const float* W    = (const float*)d_in[4];  // W         (64,2)
    const float* bias = (const float*)d_in[5];  // b         (64,)
    float* out = (float*)d_out;                 // (B,64,64,64)

    const int B = in_sizes[0] / (NCTX * 2);     // = 8
    dim3 grid(B * 16), block(256);
    convnp_fused<<<grid, block, 0, stream>>>(xc, yc, xg, sig, W, bias, out);
}